// GraphAttentionV2Layer_4045859193697
// MI455X (gfx1250) — compile-verified
//
#include <hip/hip_runtime.h>

typedef __attribute__((ext_vector_type(2))) float v2f;
typedef __attribute__((ext_vector_type(4))) float v4f;
typedef __attribute__((ext_vector_type(8))) float v8f;
typedef __attribute__((ext_vector_type(4))) int   v4i;

#define N_NODES   2048
#define IN_FEAT   128
#define N_HEADS   4
#define NHID      16
#define OUT_FEAT  64          // N_HEADS * NHID
#define NEG_SLOPE 0.2f
#define NEG_INF   (-__builtin_inff())
#define JTILE     64
#define NT        (N_NODES / JTILE)   // 32 j-tiles
#define E_STR     68                  // padded row stride for s_e (bank-conflict free)

// ---- CDNA5 async global->LDS path (ASYNCcnt), with safe fallback ----------
#if defined(__gfx1250__) && __has_builtin(__builtin_amdgcn_global_load_async_to_lds_b128)
#define HAVE_ASYNC_LDS 1
typedef __attribute__((address_space(1))) v4i gv4i;   // global int4
typedef __attribute__((address_space(3))) v4i lv4i;   // LDS int4
#define ASYNC_COPY_B128(gptr, sptr)                                          \
    __builtin_amdgcn_global_load_async_to_lds_b128(                          \
        (gv4i*)(unsigned long long)(uintptr_t)(gptr),                        \
        (lv4i*)(unsigned)(uintptr_t)(sptr), 0, 0)
#else
#define HAVE_ASYNC_LDS 0
#endif

#if defined(__gfx1250__) && __has_builtin(__builtin_amdgcn_s_wait_asynccnt)
#define WAIT_ASYNC(n) __builtin_amdgcn_s_wait_asynccnt(n)
#elif defined(__gfx1250__)
#define WAIT_ASYNC(n) asm volatile("s_wait_asynccnt %0" ::"i"(n) : "memory")
#else
#define WAIT_ASYNC(n)
#endif

// ---------------------------------------------------------------------------
// Kernel 1: projections  g = h @ W   (fp32 WMMA, 16x16 output tile per wave)
// grid = (N/16, OUT_FEAT/16, 2)  z=0 -> (W_l, g_l), z=1 -> (W_r, g_r)
// ---------------------------------------------------------------------------
__global__ __launch_bounds__(32) void gat_proj_kernel(
    const float* __restrict__ h,
    const float* __restrict__ Wl,
    const float* __restrict__ Wr,
    float* __restrict__ gl,
    float* __restrict__ gr)
{
    const float* W = (blockIdx.z == 0) ? Wl : Wr;
    float*       G = (blockIdx.z == 0) ? gl : gr;

    const int m0 = blockIdx.x * 16;
    const int n0 = blockIdx.y * 16;
    const int l  = threadIdx.x;
    const int mi = l & 15;        // M (A) / N (B,C) index for this lane
    const int hi = l >> 4;        // lane-half selects K pair

    v8f c = {};
    #pragma unroll
    for (int k0 = 0; k0 < IN_FEAT; k0 += 4) {
        const int kb = k0 + hi * 2;
        v2f a, b;
        a.x = h[(m0 + mi) * IN_FEAT + kb + 0];
        a.y = h[(m0 + mi) * IN_FEAT + kb + 1];
        b.x = W[(kb + 0) * OUT_FEAT + n0 + mi];
        b.y = W[(kb + 1) * OUT_FEAT + n0 + mi];
        c = __builtin_amdgcn_wmma_f32_16x16x4_f32(
                false, a, false, b, (short)0, c, false, false);
    }
    #pragma unroll
    for (int v = 0; v < 8; ++v) {
        const int row = m0 + v + hi * 8;
        G[row * OUT_FEAT + n0 + mi] = c[v];
    }
}

// ---------------------------------------------------------------------------
// Kernel 2: fused masked-softmax attention (flash style, double-buffered
// async global->LDS staging).  One wave per (16-row i-tile, head).
// ---------------------------------------------------------------------------
__global__ __launch_bounds__(32) void gat_attn_kernel(
    const float* __restrict__ gl,
    const float* __restrict__ gr,
    const int*   __restrict__ adj,
    const float* __restrict__ attn_w,
    float* __restrict__ out)
{
    __shared__ __align__(16) float s_gl[2][JTILE * NHID];   // g_l tiles [jj][f]
    __shared__ __align__(16) float s_gr[2][JTILE * NHID];   // g_r tiles [jj][f]
    __shared__ __align__(16) float s_e [16 * E_STR];        // e / p tile [i][jj]
    __shared__ float s_m[16], s_s[16], s_sc[16], s_mn[16], s_red[32];

    const int l    = threadIdx.x;
    const int mi   = l & 15;            // row within i-tile
    const int hi   = l >> 4;            // lane-half
    const int i0   = blockIdx.x * 16;
    const int head = blockIdx.y;
    const int hoff = head * NHID;

    if (l < 16) { s_m[l] = NEG_INF; s_s[l] = 0.f; }

    // attention vector + this lane's g_r row, in registers as v4f[4]
    v4f wv[4], riv[4];
    #pragma unroll
    for (int q = 0; q < 4; ++q) {
        wv[q]  = ((const v4f*)attn_w)[q];
        riv[q] = ((const v4f*)(gr + (i0 + mi) * OUT_FEAT + hoff))[q];
    }

    v8f acc = {};

#if HAVE_ASYNC_LDS
    // async stage of one 64x16 fp32 tile pair: 16 b128 transfers per lane
    #define STAGE_ASYNC(JT, BUF)                                              \
        do {                                                                  \
            const int j0_ = (JT) * JTILE;                                     \
            _Pragma("unroll")                                                 \
            for (int c = 0; c < 8; ++c) {                                     \
                const int t  = c * 32 + l;       /* chunk 0..255 */           \
                const int jj = t >> 2, f0 = (t & 3) << 2;                     \
                ASYNC_COPY_B128(gl + (j0_ + jj) * OUT_FEAT + hoff + f0,       \
                                &s_gl[BUF][jj * NHID + f0]);                  \
                ASYNC_COPY_B128(gr + (j0_ + jj) * OUT_FEAT + hoff + f0,       \
                                &s_gr[BUF][jj * NHID + f0]);                  \
            }                                                                 \
            asm volatile("" ::: "memory");                                    \
        } while (0)

    STAGE_ASYNC(0, 0);      // prologue: tile 0 -> buffer 0
#endif

    for (int jt = 0; jt < NT; ++jt) {
        const int cur = jt & 1;
        const int j0  = jt * JTILE;

#if HAVE_ASYNC_LDS
        if (jt + 1 < NT) {
            STAGE_ASYNC(jt + 1, cur ^ 1);   // prefetch next tile
            WAIT_ASYNC(16);                 // current tile's 16 copies done
        } else {
            WAIT_ASYNC(0);
        }
#else
        for (int t = l; t < JTILE * NHID; t += 32) {
            const int jj = t >> 4, f = t & 15;
            s_gl[cur][t] = gl[(j0 + jj) * OUT_FEAT + hoff + f];
            s_gr[cur][t] = gr[(j0 + jj) * OUT_FEAT + hoff + f];
        }
#endif
        __syncthreads();

        // prefetch next adjacency row segment (global_prefetch_b8)
        if (jt + 1 < NT)
            __builtin_prefetch(adj + (i0 + mi) * N_NODES + j0 + JTILE, 0, 1);

        // masked scores: lane (mi, hi) covers jj = hi, hi+2, ...
        const float* glbuf = s_gl[cur];
        for (int jj = hi; jj < JTILE; jj += 2) {
            const v4f* glrow = (const v4f*)&glbuf[jj * NHID];
            float e = 0.f;
            #pragma unroll
            for (int q = 0; q < 4; ++q) {
                v4f s = riv[q] + glrow[q];
                s.x = (s.x > 0.f) ? s.x : s.x * NEG_SLOPE;
                s.y = (s.y > 0.f) ? s.y : s.y * NEG_SLOPE;
                s.z = (s.z > 0.f) ? s.z : s.z * NEG_SLOPE;
                s.w = (s.w > 0.f) ? s.w : s.w * NEG_SLOPE;
                e = fmaf(s.x, wv[q].x, e);
                e = fmaf(s.y, wv[q].y, e);
                e = fmaf(s.z, wv[q].z, e);
                e = fmaf(s.w, wv[q].w, e);
            }
            if (adj[(i0 + mi) * N_NODES + j0 + jj] == 0) e = NEG_INF;
            s_e[mi * E_STR + jj] = e;
        }
        __syncthreads();

        // per-row max, all 32 lanes (each owns a 32-element half-row)
        {
            const v4f* erow = (const v4f*)&s_e[mi * E_STR + hi * 32];
            float tm = NEG_INF;
            #pragma unroll
            for (int q = 0; q < 8; ++q) {
                v4f ee = erow[q];
                tm = fmaxf(tm, fmaxf(fmaxf(ee.x, ee.y), fmaxf(ee.z, ee.w)));
            }
            s_red[hi * 16 + mi] = tm;
        }
        __syncthreads();

        if (l < 16) {
            const float mold = s_m[l];
            const float mnew = fmaxf(mold, fmaxf(s_red[l], s_red[16 + l]));
            s_mn[l] = mnew;
            s_sc[l] = (mold == NEG_INF) ? 0.f : __expf(mold - mnew);
            s_m[l]  = mnew;
        }
        __syncthreads();

        // convert e -> p = exp(e - mnew), all 32 lanes, accumulate half-row sums
        {
            const float mnew = s_mn[mi];
            v4f* erow = (v4f*)&s_e[mi * E_STR + hi * 32];
            float ps = 0.f;
            #pragma unroll
            for (int q = 0; q < 8; ++q) {
                v4f ee = erow[q], p;
                p.x = (ee.x == NEG_INF) ? 0.f : __expf(ee.x - mnew);
                p.y = (ee.y == NEG_INF) ? 0.f : __expf(ee.y - mnew);
                p.z = (ee.z == NEG_INF) ? 0.f : __expf(ee.z - mnew);
                p.w = (ee.w == NEG_INF) ? 0.f : __expf(ee.w - mnew);
                erow[q] = p;
                ps += p.x + p.y + p.z + p.w;
            }
            s_red[hi * 16 + mi] = ps;
        }
        __syncthreads();

        if (l < 16)
            s_s[l] = s_s[l] * s_sc[l] + s_red[l] + s_red[16 + l];

        // rescale accumulator rows, then  acc += P(16x64) @ Gr(64x16)  via WMMA
        #pragma unroll
        for (int v = 0; v < 8; ++v) acc[v] *= s_sc[v + hi * 8];

        const float* grbuf = s_gr[cur];
        #pragma unroll
        for (int kk = 0; kk < JTILE; kk += 4) {
            const int kb = kk + hi * 2;
            v2f a, b;
            a.x = s_e[mi * E_STR + kb + 0];
            a.y = s_e[mi * E_STR + kb + 1];
            b.x = grbuf[(kb + 0) * NHID + mi];
            b.y = grbuf[(kb + 1) * NHID + mi];
            acc = __builtin_amdgcn_wmma_f32_16x16x4_f32(
                      false, a, false, b, (short)0, acc, false, false);
        }
        __syncthreads();
    }

    // normalize (nan_to_num: fully-masked rows -> 0) and store
    if (l < 16) s_sc[l] = (s_s[l] > 0.f) ? (1.f / s_s[l]) : 0.f;
    __syncthreads();

    #pragma unroll
    for (int v = 0; v < 8; ++v) {
        const int row = v + hi * 8;
        out[(i0 + row) * OUT_FEAT + hoff + mi] = acc[v] * s_sc[row];
    }
}

// ---------------------------------------------------------------------------
extern "C" void kernel_launch(void* const* d_in, const int* in_sizes, int n_in,
                              void* d_out, int out_size, void* d_ws, size_t ws_size,
                              hipStream_t stream)
{
    const float* h   = (const float*)d_in[0];
    const int*   adj = (const int*)  d_in[1];
    const float* Wl  = (const float*)d_in[2];
    const float* Wr  = (const float*)d_in[3];
    const float* aw  = (const float*)d_in[4];
    float* out = (float*)d_out;

    float* gl = (float*)d_ws;                 // 2048*64 fp32
    float* gr = gl + N_NODES * OUT_FEAT;      // 2048*64 fp32  (1 MB total)

    dim3 g1(N_NODES / 16, OUT_FEAT / 16, 2);
    gat_proj_kernel<<<g1, 32, 0, stream>>>(h, Wl, Wr, gl, gr);

    dim3 g2(N_NODES / 16, N_HEADS);
    gat_attn_kernel<<<g2, 32, 0, stream>>>(gl, gr, adj, aw, out);
}